// MultiHeadAttentionLayer_59579786330257
// MI455X (gfx1250) — compile-verified
//
#include <hip/hip_runtime.h>
#include <math.h>

// ---------------------------------------------------------------------------
// Edge-gated multi-head graph attention (GraphTransformer layer) for gfx1250.
// FP32 everywhere, matrix work on V_WMMA_F32_16X16X4_F32 (exact fp32 GEMM).
// M-tile = 32 rows/block so each B fragment feeds 2 WMMAs (edge kernel) or
// 6 WMMAs per A/B fetch group (node kernel) -> better compute/load ratio.
// ---------------------------------------------------------------------------

typedef __attribute__((ext_vector_type(2))) float v2f;
typedef __attribute__((ext_vector_type(8))) float v8f;

#define IN_DIM   128
#define HD       128   // NUM_HEAD * OUT_DIM
#define NUM_HEAD 8
#define OUT_DIM  16
#define LDS_STR  132   // 128 + 4 pad: stride%64==4 -> conflict-free fragment reads
#define MTILE    32    // rows (nodes/edges) staged per block

__device__ __forceinline__ v8f wmma_f32(v2f a, v2f b, v8f c) {
    // (neg_a, A, neg_b, B, c_mod, C, reuse_a, reuse_b)
    return __builtin_amdgcn_wmma_f32_16x16x4_f32(false, a, false, b, (short)0, c,
                                                 false, false);
}

// Cooperative stage of a MTILE x 128 fp32 tile into padded LDS, float4-wide.
__device__ __forceinline__ void stage_tile(float* __restrict__ S,
                                           const float* __restrict__ G,
                                           int r0, int n_rows, int tid) {
    for (int i = tid; i < MTILE * (IN_DIM / 4); i += 256) {
        const int r  = i >> 5;           // row in tile
        const int c4 = i & 31;           // float4 column
        const int row = r0 + r;
        float4 v = make_float4(0.f, 0.f, 0.f, 0.f);
        if (row < n_rows)
            v = ((const float4*)&G[(size_t)row * IN_DIM])[c4];
        ((float4*)&S[r * LDS_STR])[c4] = v;
    }
}

// ---------------------------------------------------------------------------
// Kernel 0: zero the atomic accumulation buffers (d_out and wZ).
// ---------------------------------------------------------------------------
__global__ void ga_zero_kernel(float* __restrict__ out, float* __restrict__ accZ,
                               int n_out, int n_z) {
    int i = blockIdx.x * blockDim.x + threadIdx.x;
    if (i < n_out) out[i] = 0.0f;
    if (i < n_z)   accZ[i] = 0.0f;
}

// ---------------------------------------------------------------------------
// Kernel 1: node projections Qh/Kh/Vh = X @ W{Q,K,V} + b   ([N,128]x[128,128])
// One block = 32 node rows; wave w owns column slab w*16 and carries six
// accumulators (2 M-tiles x {Q,K,V}) -> 192 v_wmma per wave, 6 per B-triple.
// ---------------------------------------------------------------------------
__global__ __launch_bounds__(256) void ga_node_proj_kernel(
    const float* __restrict__ X,
    const float* __restrict__ WQ, const float* __restrict__ bQ,
    const float* __restrict__ WK, const float* __restrict__ bK,
    const float* __restrict__ WV, const float* __restrict__ bV,
    float* __restrict__ Qh, float* __restrict__ Kh, float* __restrict__ Vh,
    int n_nodes)
{
    __shared__ __align__(16) float As[MTILE * LDS_STR];
    const int tid = threadIdx.x;
    const int m0  = blockIdx.x * MTILE;

    stage_tile(As, X, m0, n_nodes, tid);
    __syncthreads();

    const int wave = tid >> 5;
    const int lane = tid & 31;
    const int lrow = lane & 15;   // A row within 16-tile / D column
    const int half = lane >> 4;   // K-half selector for 32-bit A layout
    const int n0   = wave * 16;   // this wave's output column slab

    v8f accQ[2] = {{}, {}};
    v8f accK[2] = {{}, {}};
    v8f accV[2] = {{}, {}};

    #pragma unroll 4
    for (int k = 0; k < IN_DIM; k += 4) {
        const int kk = k + 2 * half;   // lanes 0-15 -> K{k,k+1}, 16-31 -> K{k+2,k+3}
        v2f a0, a1;
        a0.x = As[lrow * LDS_STR + kk];
        a0.y = As[lrow * LDS_STR + kk + 1];
        a1.x = As[(16 + lrow) * LDS_STR + kk];
        a1.y = As[(16 + lrow) * LDS_STR + kk + 1];

        const int bi = kk * HD + n0 + lrow;   // W row-major [128,128]
        v2f b0, b1, b2;
        b0.x = WQ[bi]; b0.y = WQ[bi + HD];
        b1.x = WK[bi]; b1.y = WK[bi + HD];
        b2.x = WV[bi]; b2.y = WV[bi + HD];

        accQ[0] = wmma_f32(a0, b0, accQ[0]);
        accQ[1] = wmma_f32(a1, b0, accQ[1]);
        accK[0] = wmma_f32(a0, b1, accK[0]);
        accK[1] = wmma_f32(a1, b1, accK[1]);
        accV[0] = wmma_f32(a0, b2, accV[0]);
        accV[1] = wmma_f32(a1, b2, accV[1]);
    }

    const float biq = bQ[n0 + lrow];
    const float bik = bK[n0 + lrow];
    const float biv = bV[n0 + lrow];
    #pragma unroll
    for (int t = 0; t < 2; ++t) {
        #pragma unroll
        for (int j = 0; j < 8; ++j) {
            int row = m0 + t * 16 + j + 8 * half;  // C/D layout: VGPR j -> M=j+8*half
            if (row < n_nodes) {
                size_t o = (size_t)row * HD + n0 + lrow;
                Qh[o] = accQ[t][j] + biq;
                Kh[o] = accK[t][j] + bik;
                Vh[o] = accV[t][j] + biv;
            }
        }
    }
}

// ---------------------------------------------------------------------------
// Kernel 2: fused edge pipeline. One block = 32 edges, wave w = head w.
//   Eh tiles (2 x 16 edges x 16 dims) via WMMA, never spilled to HBM.
//   dot = sum_d Kh[src]*Qh[dst]*Eh  (16-lane butterfly over d)
//   score = exp(dot/4);  atomic segment-sum of score*Vh[src] and score.
// ---------------------------------------------------------------------------
__global__ __launch_bounds__(256) void ga_edge_attn_kernel(
    const float* __restrict__ EF,
    const int* __restrict__ src, const int* __restrict__ dst,
    const float* __restrict__ WE, const float* __restrict__ bE,
    const float* __restrict__ Qh, const float* __restrict__ Kh,
    const float* __restrict__ Vh,
    float* __restrict__ outV,          // [N,128] atomic accumulation (d_out)
    float* __restrict__ accZ,          // [N,8]
    int n_edges)
{
    __shared__ __align__(16) float Es[MTILE * LDS_STR];
    const int tid = threadIdx.x;
    const int e0  = blockIdx.x * MTILE;

    stage_tile(Es, EF, e0, n_edges, tid);
    __syncthreads();

    const int h    = tid >> 5;    // head == wave (NUM_HEAD == waves/block)
    const int lane = tid & 31;
    const int d    = lane & 15;   // dim within head (== output column)
    const int half = lane >> 4;

    v8f acc[2] = {{}, {}};
    #pragma unroll 4
    for (int k = 0; k < IN_DIM; k += 4) {
        const int kk = k + 2 * half;
        v2f a0, a1;
        a0.x = Es[d * LDS_STR + kk];          // A row = lane&15 = d here
        a0.y = Es[d * LDS_STR + kk + 1];
        a1.x = Es[(16 + d) * LDS_STR + kk];
        a1.y = Es[(16 + d) * LDS_STR + kk + 1];
        const int bi = kk * HD + h * OUT_DIM + d;
        v2f b;
        b.x = WE[bi];
        b.y = WE[bi + HD];
        acc[0] = wmma_f32(a0, b, acc[0]);     // one B fragment feeds two WMMAs
        acc[1] = wmma_f32(a1, b, acc[1]);
    }

    const float be = bE[h * OUT_DIM + d];

    #pragma unroll
    for (int t = 0; t < 2; ++t) {
        #pragma unroll
        for (int j = 0; j < 8; ++j) {
            const int e = e0 + t * 16 + j + 8 * half;  // VGPR j holds edge j+8*half
            if (e < n_edges) {
                const int s  = src[e];
                const int dn = dst[e];
                const float eh = acc[t][j] + be;
                const size_t so = (size_t)s  * HD + h * OUT_DIM + d;
                const size_t qo = (size_t)dn * HD + h * OUT_DIM + d;
                float v = Kh[so] * Qh[qo] * eh;
                // butterfly sum over the 16 lanes holding d=0..15 for this edge
                v += __shfl_xor(v, 1, 32);
                v += __shfl_xor(v, 2, 32);
                v += __shfl_xor(v, 4, 32);
                v += __shfl_xor(v, 8, 32);
                const float score = __expf(v * 0.25f);  // 1/sqrt(OUT_DIM) = 1/4
                atomicAdd(&outV[qo], score * Vh[so]);   // coalesced 64B / 16 lanes
                if (d == 0) atomicAdd(&accZ[(size_t)dn * NUM_HEAD + h], score);
            }
        }
    }
}

// ---------------------------------------------------------------------------
// Kernel 3: out = accV / (wZ + eps), in place on d_out.
// ---------------------------------------------------------------------------
__global__ void ga_normalize_kernel(float* __restrict__ out,
                                    const float* __restrict__ accZ, int n_total) {
    int i = blockIdx.x * blockDim.x + threadIdx.x;
    if (i < n_total) {
        int n = i >> 7;              // node
        int h = (i >> 4) & 7;        // head
        out[i] = out[i] / (accZ[n * NUM_HEAD + h] + 1e-9f);
    }
}

// ---------------------------------------------------------------------------
extern "C" void kernel_launch(void* const* d_in, const int* in_sizes, int n_in,
                              void* d_out, int out_size, void* d_ws, size_t ws_size,
                              hipStream_t stream) {
    const float* node_feat = (const float*)d_in[0];
    const float* edge_feat = (const float*)d_in[1];
    const int*   src       = (const int*)d_in[2];
    const int*   dst       = (const int*)d_in[3];
    const float* WQ = (const float*)d_in[4];
    const float* bQ = (const float*)d_in[5];
    const float* WK = (const float*)d_in[6];
    const float* bK = (const float*)d_in[7];
    const float* WV = (const float*)d_in[8];
    const float* bV = (const float*)d_in[9];
    const float* WE = (const float*)d_in[10];
    const float* bE = (const float*)d_in[11];
    float* out = (float*)d_out;

    const int n_nodes = in_sizes[0] / IN_DIM;
    const int n_edges = in_sizes[2];

    // Workspace: Qh | Kh | Vh | wZ
    float* Qh   = (float*)d_ws;
    float* Kh   = Qh + (size_t)n_nodes * HD;
    float* Vh   = Kh + (size_t)n_nodes * HD;
    float* accZ = Vh + (size_t)n_nodes * HD;

    const int n_out = n_nodes * HD;
    const int n_z   = n_nodes * NUM_HEAD;

    ga_zero_kernel<<<(n_out + 255) / 256, 256, 0, stream>>>(out, accZ, n_out, n_z);

    ga_node_proj_kernel<<<(n_nodes + MTILE - 1) / MTILE, 256, 0, stream>>>(
        node_feat, WQ, bQ, WK, bK, WV, bV, Qh, Kh, Vh, n_nodes);

    ga_edge_attn_kernel<<<(n_edges + MTILE - 1) / MTILE, 256, 0, stream>>>(
        edge_feat, src, dst, WE, bE, Qh, Kh, Vh, out, accZ, n_edges);

    ga_normalize_kernel<<<(n_out + 255) / 256, 256, 0, stream>>>(out, accZ, n_out);
}